// Conv_44220983280108
// MI455X (gfx1250) — compile-verified
//
#include <hip/hip_runtime.h>

// CDNA5 (gfx1250) implicit-GEMM 3x3 SAME conv + ReLU via V_WMMA_F32_16X16X4_F32.
// x: [224,224,128] f32   W: [256, 1152] f32 (flat k = kh*3*128 + kw*128 + c)
// y: [224,224,256] f32
//
// Data path:
//   - Weights repacked once per call into d_ws as Wp[k/2][n][2] so the WMMA B
//     operand is a coalesced 8B/lane load with pure immediate-offset addressing.
//   - Activations staged per block into LDS (3-row halo slab, 528B pixel stride
//     => conflict-free ds_load_b64 for the WMMA A operand; zero slot for padding).

typedef float v2f __attribute__((ext_vector_type(2)));
typedef float v8f __attribute__((ext_vector_type(8)));

#define IH   224
#define IW   224
#define PIX  (IH * IW)   // 50176
#define CIN  128
#define COUT 256
#define KTOT 1152        // 9 * 128

#define TILE_P   32      // pixels per block (2 M-tiles)
#define JDIM     34      // staged pixels per band: TILE_P + 2 halo
#define PSTRIDE  132     // floats per staged pixel (128 + 4 pad -> 528B, bank-conflict-free)
#define BANDF    (JDIM * PSTRIDE)       // 4488 floats per band
#define ZBASE    (3 * BANDF)            // 13464: zero slot for padding taps
#define SLABF    (ZBASE + PSTRIDE)      // 13596 floats = 54384 B LDS

#define WP_FLOATS (KTOT * COUT)         // 294912 floats = 1.18 MB repacked weights

// ---- Pre-pass: W[n][k] -> Wp[k/2][n][k%2]  (one float per thread)
__global__ __launch_bounds__(256)
void repack_w_kernel(const float* __restrict__ w, float* __restrict__ wp) {
    const int o  = blockIdx.x * 256 + threadIdx.x;   // 0 .. 294911
    const int k2 = o >> 9;         // o / 512
    const int r  = o & 511;
    const int n  = r >> 1;
    const int p  = r & 1;
    wp[o] = w[n * KTOT + k2 * 2 + p];
}

template <bool REPACKED>
__global__ __launch_bounds__(128)
void conv3x3_wmma_f32(const float* __restrict__ x,
                      const float* __restrict__ wB,   // repacked (d_ws) or original W
                      float* __restrict__ y) {
    __shared__ float slab[SLABF];

    const int tid  = threadIdx.x;       // 0..127 (4 waves)
    const int lane = tid & 31;
    const int wave = tid >> 5;
    const int l15  = lane & 15;         // M (A) / N (B,C,D)
    const int ksel = lane >> 4;         // selects K pair {0,1} vs {2,3}

    const int p0 = blockIdx.x * TILE_P; // first flat output pixel of this block
    const int n0 = wave * 64;           // first output channel of this wave (4 N-tiles)

    // ---------------- Stage activation slab into LDS ----------------
    if (tid < PSTRIDE) slab[ZBASE + tid] = 0.0f;   // zero slot for padding taps

    #pragma unroll
    for (int band = 0; band < 3; ++band) {
        const int sd = p0 + (band - 1) * IW - 1;   // first staged flat pixel of band
        for (int e = tid; e < JDIM * (CIN / 4); e += 128) {
            const int j  = e >> 5;                 // staged pixel within band, 0..33
            const int c4 = (e & 31) << 2;          // channel quad, 0..124
            int q = sd + j;                        // clamp only to stay in-bounds;
            q = q < 0 ? 0 : (q > PIX - 1 ? PIX - 1 : q);  // garbage masked at consume
            const float4 v = *(const float4*)(x + (size_t)q * CIN + c4);
            *(float4*)&slab[band * BANDF + j * PSTRIDE + c4] = v;
        }
    }
    __syncthreads();

    // ---------------- Per-lane B base (coalesced, immediate-offset K walk) ----
    // Repacked: float index = (k/2 + ksel)*2*COUT + n*2 + p
    const float* bbase[4];
    #pragma unroll
    for (int nt = 0; nt < 4; ++nt) {
        const int n = n0 + nt * 16 + l15;
        bbase[nt] = REPACKED ? (wB + ksel * (2 * COUT) + n * 2)
                             : (wB + (size_t)n * KTOT + ksel * 2);
    }

    v8f acc[2][4];
    #pragma unroll
    for (int mt = 0; mt < 2; ++mt)
        #pragma unroll
        for (int nt = 0; nt < 4; ++nt)
            acc[mt][nt] = {};

    // Per-lane output pixel coordinates per M-tile.
    int hh[2], ww[2];
    #pragma unroll
    for (int mt = 0; mt < 2; ++mt) {
        const int p = p0 + mt * 16 + l15;
        hh[mt] = p / IW;
        ww[mt] = p % IW;
    }

    // ---------------- K loop: 9 taps x 32 chunks of K=4 ----------------
    for (int kk = 0; kk < 9; ++kk) {
        const int kh = kk / 3 - 1;
        const int kw = kk % 3 - 1;

        // LDS float offsets for the A operand of each M-tile (zero slot if OOB).
        int aoff[2];
        #pragma unroll
        for (int mt = 0; mt < 2; ++mt) {
            const int h2 = hh[mt] + kh;
            const int w2 = ww[mt] + kw;
            const bool valid = ((unsigned)h2 < (unsigned)IH) &&
                               ((unsigned)w2 < (unsigned)IW);
            const int j = mt * 16 + l15 + kw + 1;              // 0..33
            aoff[mt] = valid ? ((kh + 1) * BANDF + j * PSTRIDE + ksel * 2)
                             : (ZBASE + ksel * 2);
        }

        #pragma unroll 4
        for (int c = 0; c < CIN; c += 4) {
            v2f a[2];
            #pragma unroll
            for (int mt = 0; mt < 2; ++mt)
                a[mt] = *(const v2f*)&slab[aoff[mt] + c];       // ds_load_b64

            v2f b[4];
            #pragma unroll
            for (int nt = 0; nt < 4; ++nt)
                b[nt] = REPACKED
                    ? *(const v2f*)(bbase[nt] + kk * (64 * 2 * COUT) + c * (COUT / 2) * 4)
                    : *(const v2f*)(bbase[nt] + kk * CIN + c);

            #pragma unroll
            for (int mt = 0; mt < 2; ++mt)
                #pragma unroll
                for (int nt = 0; nt < 4; ++nt)
                    acc[mt][nt] = __builtin_amdgcn_wmma_f32_16x16x4_f32(
                        false, a[mt], false, b[nt], (short)0, acc[mt][nt],
                        false, false);
        }
    }

    // ---------------- Epilogue: ReLU + store ----------------
    // D layout: VGPR r holds M = r + 8*ksel, lane l15 = N.
    float* ybase = y + (size_t)(p0 + ksel * 8) * COUT + n0 + l15;
    #pragma unroll
    for (int mt = 0; mt < 2; ++mt)
        #pragma unroll
        for (int nt = 0; nt < 4; ++nt)
            #pragma unroll
            for (int r = 0; r < 8; ++r) {
                float v = acc[mt][nt][r];
                v = v > 0.0f ? v : 0.0f;
                ybase[(size_t)(mt * 16 + r) * COUT + nt * 16] = v;
            }
}

extern "C" void kernel_launch(void* const* d_in, const int* in_sizes, int n_in,
                              void* d_out, int out_size, void* d_ws, size_t ws_size,
                              hipStream_t stream) {
    const float* x = (const float*)d_in[0];  // 224*224*128
    const float* w = (const float*)d_in[1];  // 256*1152
    float* y       = (float*)d_out;          // 224*224*256

    const int blocks = PIX / TILE_P;         // 1568

    if (ws_size >= (size_t)WP_FLOATS * sizeof(float)) {
        float* wp = (float*)d_ws;
        hipLaunchKernelGGL(repack_w_kernel, dim3(WP_FLOATS / 256), dim3(256), 0,
                           stream, w, wp);
        hipLaunchKernelGGL((conv3x3_wmma_f32<true>), dim3(blocks), dim3(128), 0,
                           stream, x, wp, y);
    } else {
        hipLaunchKernelGGL((conv3x3_wmma_f32<false>), dim3(blocks), dim3(128), 0,
                           stream, x, w, y);
    }
}